// MoEHook_77549929496922
// MI455X (gfx1250) — compile-verified
//
#include <hip/hip_runtime.h>
#include <hip/hip_bf16.h>

typedef __attribute__((ext_vector_type(2))) float v2f;
typedef __attribute__((ext_vector_type(8))) float v8f;

#define T_TOK 4096
#define H_DIM 1024
#define F_DIM 4096
#define N_EXP 8

#define BM 32
#define BN 128
#define KB 32
#define XS_STRIDE 36   // 144 B rows: 16B aligned, distinct banks mod 64
#define WS_STRIDE 132  // 528 B rows: 16B aligned

// ---- CDNA5 async global->LDS path (probe-guarded) -------------------------
#if defined(__has_builtin)
#if __has_builtin(__builtin_amdgcn_global_load_async_to_lds_b128)
#define HAVE_ASYNC_LDS 1
#endif
#endif
#ifndef HAVE_ASYNC_LDS
#define HAVE_ASYNC_LDS 0
#endif

#if defined(__has_builtin) && __has_builtin(__builtin_amdgcn_s_wait_asynccnt)
#define WAIT_ASYNC(n) __builtin_amdgcn_s_wait_asynccnt(n)
#else
#define WAIT_ASYNC(n) asm volatile("s_wait_asynccnt %0" ::"i"(n) : "memory")
#endif

// payload type for the async b128 copy: 16-byte int vector,
// src in global (AS1, prints as __device__), dst in LDS (AS3, __shared__)
typedef int v4i_t __attribute__((vector_size(16)));
typedef __attribute__((address_space(1))) v4i_t* g4ptr_t;
typedef __attribute__((address_space(3))) v4i_t* l4ptr_t;

// ---------------------------------------------------------------------------
// Router: one wave per token. logits = x @ router_w, softmax+top2+renorm.
// ---------------------------------------------------------------------------
__global__ __launch_bounds__(256)
void router_kernel(const float* __restrict__ x,
                   const float* __restrict__ rw,
                   float* __restrict__ combine) {
    const int lane = threadIdx.x & 31;
    const int wave = threadIdx.x >> 5;
    const int t = blockIdx.x * 8 + wave;
    if (t >= T_TOK) return;

    float acc[N_EXP];
#pragma unroll
    for (int e = 0; e < N_EXP; ++e) acc[e] = 0.0f;

    const float* xr = x + (size_t)t * H_DIM;
    for (int h = lane; h < H_DIM; h += 32) {
        const float xv = xr[h];
#pragma unroll
        for (int e = 0; e < N_EXP; ++e) acc[e] += xv * rw[h * N_EXP + e];
    }
#pragma unroll
    for (int e = 0; e < N_EXP; ++e) {
#pragma unroll
        for (int off = 16; off > 0; off >>= 1)
            acc[e] += __shfl_xor(acc[e], off, 32);
    }
    if (lane == 0) {
        int i1 = 0; float v1 = acc[0];
#pragma unroll
        for (int e = 1; e < N_EXP; ++e)
            if (acc[e] > v1) { v1 = acc[e]; i1 = e; }
        int i2 = (i1 == 0) ? 1 : 0; float v2 = acc[i2];
#pragma unroll
        for (int e = 0; e < N_EXP; ++e)
            if (e != i1 && acc[e] > v2) { v2 = acc[e]; i2 = e; }
        // renormalized top-2 of softmax == softmax over the two largest logits
        const float w1 = 1.0f / (1.0f + __expf(v2 - v1));
        const float w2 = 1.0f - w1;
        float* cr = combine + (size_t)t * N_EXP;
#pragma unroll
        for (int e = 0; e < N_EXP; ++e) cr[e] = 0.0f;
        cr[i1] = w1;
        cr[i2] = w2;
    }
}

// ---------------------------------------------------------------------------
// WMMA fp32 GEMM: C[M,N] (+)= post( A[M,K] @ B[K,N] )
// Double-buffered LDS staging via async global->LDS copies (ASYNCcnt),
// two 16x16 f32 accumulators per wave, V_WMMA_F32_16X16X4_F32 inner loop.
// ---------------------------------------------------------------------------
__global__ __launch_bounds__(256)
void gemm_wmma(const float* __restrict__ A,
               const float* __restrict__ B,
               float* __restrict__ C,
               int N, int K,
               int act_silu,
               const float* __restrict__ combine, int expert,
               int accumulate) {
    __shared__ float Xs[2][BM][XS_STRIDE];
    __shared__ float Ws[2][KB][WS_STRIDE];

    const int tid   = threadIdx.x;
    const int lane  = tid & 31;
    const int wave  = tid >> 5;
    const int row0  = blockIdx.y * BM;
    const int col0  = blockIdx.x * BN;
    const int klane = lane >> 4;   // 0: K pair {0,1}, 1: K pair {2,3}
    const int nlane = lane & 15;   // M (A frag) / N (B,C frags)

    // per-thread staging coordinates
    const int ar = (tid * 4) >> 5;        // A tile: 32x32, one b128 per thread
    const int ac = (tid * 4) & 31;

    auto stage = [&](int buf, int kb) {
#if HAVE_ASYNC_LDS
        __builtin_amdgcn_global_load_async_to_lds_b128(
            (g4ptr_t)(A + (size_t)(row0 + ar) * K + kb + ac),
            (l4ptr_t)&Xs[buf][ar][ac], 0, 0);
#pragma unroll
        for (int j = 0; j < 4; ++j) {
            const int idx = (tid + j * 256) * 4;
            const int r = idx >> 7;
            const int c = idx & 127;
            __builtin_amdgcn_global_load_async_to_lds_b128(
                (g4ptr_t)(B + (size_t)(kb + r) * N + col0 + c),
                (l4ptr_t)&Ws[buf][r][c], 0, 0);
        }
#else
        {
            const float4 v =
                *(const float4*)(A + (size_t)(row0 + ar) * K + kb + ac);
            *(float4*)&Xs[buf][ar][ac] = v;
        }
#pragma unroll
        for (int j = 0; j < 4; ++j) {
            const int idx = (tid + j * 256) * 4;
            const int r = idx >> 7;
            const int c = idx & 127;
            const float4 v =
                *(const float4*)(B + (size_t)(kb + r) * N + col0 + c);
            *(float4*)&Ws[buf][r][c] = v;
        }
#endif
    };

    v8f acc0 = {};
    v8f acc1 = {};

    const int nchunks = K / KB;
    stage(0, 0);
    if (nchunks > 1) stage(1, KB);

    const int nw = wave * 16 + nlane;
    for (int i = 0; i < nchunks; ++i) {
        const int p = i & 1;
#if HAVE_ASYNC_LDS
        // 5 async copies per chunk per thread; keep the next chunk in flight
        if (i + 1 < nchunks) WAIT_ASYNC(5);
        else                 WAIT_ASYNC(0);
#endif
        __syncthreads();  // all waves' copies for buffer p have landed

#pragma unroll
        for (int k4 = 0; k4 < KB; k4 += 4) {
            const int kk = k4 + klane * 2;
            v2f b;
            b.x = Ws[p][kk][nw];
            b.y = Ws[p][kk + 1][nw];
            const v2f a0 = *(const v2f*)&Xs[p][nlane][kk];
            const v2f a1 = *(const v2f*)&Xs[p][16 + nlane][kk];
            acc0 = __builtin_amdgcn_wmma_f32_16x16x4_f32(
                false, a0, false, b, (short)0, acc0, false, false);
            acc1 = __builtin_amdgcn_wmma_f32_16x16x4_f32(
                false, a1, false, b, (short)0, acc1, false, false);
        }
        __syncthreads();  // everyone done reading buffer p before restaging it

        if (i + 2 < nchunks) stage(p, (i + 2) * KB);
    }

    // epilogue: C/D layout -> element (r + 8*klane, nlane) per VGPR r
    const int cg = col0 + wave * 16 + nlane;
#pragma unroll
    for (int r = 0; r < 8; ++r) {
        const int gr0 = row0 + r + 8 * klane;
        const int gr1 = gr0 + 16;
        float v0 = acc0[r];
        float v1 = acc1[r];
        if (act_silu) {
            v0 = v0 / (1.0f + __expf(-v0));
            v1 = v1 / (1.0f + __expf(-v1));
        }
        if (combine) {
            v0 *= combine[gr0 * N_EXP + expert];
            v1 *= combine[gr1 * N_EXP + expert];
        }
        const size_t o0 = (size_t)gr0 * N + cg;
        const size_t o1 = (size_t)gr1 * N + cg;
        if (accumulate) {
            C[o0] += v0;
            C[o1] += v1;
        } else {
            C[o0] = v0;
            C[o1] = v1;
        }
    }
}

// ---------------------------------------------------------------------------
extern "C" void kernel_launch(void* const* d_in, const int* in_sizes, int n_in,
                              void* d_out, int out_size, void* d_ws, size_t ws_size,
                              hipStream_t stream) {
    const float* x        = (const float*)d_in[0];  // [B,S,H] -> [T,H]
    const float* router_w = (const float*)d_in[1];  // [H,E]
    const float* base_w1  = (const float*)d_in[2];  // [H,F]
    const float* base_w2  = (const float*)d_in[3];  // [F,H]
    const float* exp_w1   = (const float*)d_in[4];  // [E,H,F]
    const float* exp_w2   = (const float*)d_in[5];  // [E,F,H]
    float* out            = (float*)d_out;          // [T,H]

    float* mid     = (float*)d_ws;                                    // [T,F]
    float* combine = (float*)((char*)d_ws +
                              (size_t)T_TOK * F_DIM * sizeof(float)); // [T,E]

    // 1) router -> dense combine weights
    router_kernel<<<T_TOK / 8, 256, 0, stream>>>(x, router_w, combine);

    const dim3 blk(256);
    const dim3 g1(F_DIM / BN, T_TOK / BM);  // GEMM1: N=F, K=H
    const dim3 g2(H_DIM / BN, T_TOK / BM);  // GEMM2: N=H, K=F

    // 2) base FFN: out = silu(x @ w1) @ w2
    gemm_wmma<<<g1, blk, 0, stream>>>(x, base_w1, mid, F_DIM, H_DIM,
                                      /*silu=*/1, nullptr, 0, /*acc=*/0);
    gemm_wmma<<<g2, blk, 0, stream>>>(mid, base_w2, out, H_DIM, F_DIM,
                                      /*silu=*/0, nullptr, 0, /*acc=*/0);

    // 3) experts: out += (combine[:,e] * silu(x @ w1_e)) @ w2_e
    for (int e = 0; e < N_EXP; ++e) {
        gemm_wmma<<<g1, blk, 0, stream>>>(
            x, exp_w1 + (size_t)e * H_DIM * F_DIM, mid, F_DIM, H_DIM,
            /*silu=*/1, combine, e, /*acc=*/0);
        gemm_wmma<<<g2, blk, 0, stream>>>(
            mid, exp_w2 + (size_t)e * F_DIM * H_DIM, out, H_DIM, F_DIM,
            /*silu=*/0, nullptr, 0, /*acc=*/1);
    }
}